// DenseGATLayer_90108413870812
// MI455X (gfx1250) — compile-verified
//
#include <hip/hip_runtime.h>
#include <cstdint>
#include <cstddef>

#define N_NODES  2048
#define IN_SIZE  256
#define HEADS    4
#define OUT_SIZE 32
#define FEAT_COLS (HEADS * OUT_SIZE)   // 128

typedef float v2f __attribute__((ext_vector_type(2)));
typedef float v4f __attribute__((ext_vector_type(4)));
typedef float v8f __attribute__((ext_vector_type(8)));

// ---------------------------------------------------------------------------
// Kernel 1: feat = h @ W^T   [2048,256] x [256,128] -> [2048,128], fp32 WMMA.
// One wave computes a 16-row strip x 128 columns (8 tiles of 16x16).
// V_WMMA_F32_16X16X4_F32: A is 16x4 (2 VGPRs/lane), B is 4x16, C/D 16x16 (8 VGPRs).
// A layout: lanes 0-15 -> M=lane, K={k0,k0+1}; lanes 16-31 -> M=lane-16, K={k0+2,k0+3}.
// B layout mirrors with N across lanes.
// ---------------------------------------------------------------------------
__global__ __launch_bounds__(128) void gat_feat_wmma(
    const float* __restrict__ h, const float* __restrict__ W,
    float* __restrict__ feat)
{
  const int lane  = threadIdx.x & 31;
  const int wave  = threadIdx.x >> 5;
  const int m0    = blockIdx.x * 64 + wave * 16;   // row strip base
  const int khalf = lane >> 4;                     // 0: K pair {0,1}, 1: {2,3}
  const int lid   = lane & 15;

  v8f acc[8];
#pragma unroll
  for (int c = 0; c < 8; ++c) acc[c] = (v8f)(0.0f);

  const float* __restrict__ arow = h + (size_t)(m0 + lid) * IN_SIZE;

  for (int k0 = 0; k0 < IN_SIZE; k0 += 4) {
    const int ka = k0 + khalf * 2;
    v2f a;
    a.x = arow[ka];
    a.y = arow[ka + 1];
#pragma unroll
    for (int c = 0; c < 8; ++c) {
      // B[k][n] = (W^T)[k][n] = W[n][k], n = c*16 + lid
      const float* __restrict__ brow = W + (size_t)(c * 16 + lid) * IN_SIZE;
      v2f b;
      b.x = brow[ka];
      b.y = brow[ka + 1];
      acc[c] = __builtin_amdgcn_wmma_f32_16x16x4_f32(
          /*neg_a=*/false, a, /*neg_b=*/false, b,
          /*c_mod=*/(short)0, acc[c], /*reuse_a=*/false, /*reuse_b=*/false);
    }
  }

  // C/D layout: VGPR r -> (M = m0 + r + khalf*8, N = c*16 + lid)
#pragma unroll
  for (int c = 0; c < 8; ++c) {
#pragma unroll
    for (int r = 0; r < 8; ++r) {
      const int m = m0 + r + khalf * 8;
      const int n = c * 16 + lid;
      feat[(size_t)m * FEAT_COLS + n] = acc[c][r];
    }
  }
}

// ---------------------------------------------------------------------------
// Kernel 2: per-row attention weights + head-collapse.
//   s[n,h]  = sum_o feat[n,h,o] * (a[h,o] + a[h,o+32])
//   e       = leaky_relu(2*s, 0.01)
//   p[n,h]  = softmax over the 4 heads
//   v[n,o]  = sum_h p[n,h] * feat[n,h,o]
// One wave (32 lanes == OUT_SIZE) per row n.
// ---------------------------------------------------------------------------
__global__ __launch_bounds__(256) void gat_attn_reduce(
    const float* __restrict__ feat, const float* __restrict__ attn_a,
    float* __restrict__ v)
{
  const int lane = threadIdx.x & 31;   // == output channel o
  const int wave = threadIdx.x >> 5;
  const int n    = blockIdx.x * 8 + wave;

  const float* __restrict__ f = feat + (size_t)n * FEAT_COLS;

  float fh[HEADS], t[HEADS];
#pragma unroll
  for (int hd = 0; hd < HEADS; ++hd) {
    fh[hd] = f[hd * OUT_SIZE + lane];
    const float a = attn_a[hd * 2 * OUT_SIZE + lane] +
                    attn_a[hd * 2 * OUT_SIZE + OUT_SIZE + lane];
    t[hd] = fh[hd] * a;
  }

  // full-wave (32-lane) butterfly reduction for each head
#pragma unroll
  for (int off = 16; off > 0; off >>= 1) {
#pragma unroll
    for (int hd = 0; hd < HEADS; ++hd)
      t[hd] += __shfl_xor(t[hd], off, 32);
  }

  float e[HEADS];
#pragma unroll
  for (int hd = 0; hd < HEADS; ++hd) {
    const float x = 2.0f * t[hd];
    e[hd] = x > 0.0f ? x : 0.01f * x;
  }
  const float mx = fmaxf(fmaxf(e[0], e[1]), fmaxf(e[2], e[3]));
  float p[HEADS], s = 0.0f;
#pragma unroll
  for (int hd = 0; hd < HEADS; ++hd) {
    p[hd] = __expf(e[hd] - mx);
    s += p[hd];
  }
  const float inv = 1.0f / s;

  float vo = 0.0f;
#pragma unroll
  for (int hd = 0; hd < HEADS; ++hd) vo += (p[hd] * inv) * fh[hd];

  v[(size_t)n * OUT_SIZE + lane] = vo;
}

// ---------------------------------------------------------------------------
// Kernel 3: out[n, j, :] = adj[n,j] > 0 ? v[n,:] : NaN   (512 MB streaming)
// 256 threads: 32 j's per block, 8 threads (float4 each) cover the 32 channels.
// Non-temporal B128 stores: output has zero reuse and exceeds L2.
// ---------------------------------------------------------------------------
__global__ __launch_bounds__(256) void gat_broadcast(
    const float* __restrict__ adj, const float* __restrict__ v,
    float* __restrict__ out)
{
  const int n  = blockIdx.y;
  const int j  = blockIdx.x * 32 + (threadIdx.x >> 3);
  const int o4 = (threadIdx.x & 7) * 4;

  const v4f* __restrict__ vrow = (const v4f*)(v + (size_t)n * OUT_SIZE);
  v4f val = vrow[o4 >> 2];

  const float a = adj[(size_t)n * N_NODES + j];
  if (!(a > 0.0f)) {
    const float qnan = __builtin_nanf("");
    val = (v4f){qnan, qnan, qnan, qnan};
  }

  v4f* dst = (v4f*)(out + ((size_t)n * N_NODES + j) * OUT_SIZE + o4);
  __builtin_nontemporal_store(val, dst);
}

extern "C" void kernel_launch(void* const* d_in, const int* in_sizes, int n_in,
                              void* d_out, int out_size, void* d_ws, size_t ws_size,
                              hipStream_t stream) {
  (void)in_sizes; (void)n_in; (void)out_size; (void)ws_size;

  const float* adj    = (const float*)d_in[0];  // [2048, 2048]
  const float* h      = (const float*)d_in[1];  // [2048, 256]
  const float* W      = (const float*)d_in[2];  // [128, 256]
  const float* attn_a = (const float*)d_in[3];  // [1, 4, 64]
  float* out = (float*)d_out;                   // [2048, 2048, 32]

  float* feat = (float*)d_ws;                         // 2048*128 fp32 = 1 MB
  float* v    = feat + (size_t)N_NODES * FEAT_COLS;   // 2048*32 fp32 = 256 KB

  gat_feat_wmma<<<dim3(N_NODES / 64), dim3(128), 0, stream>>>(h, W, feat);
  gat_attn_reduce<<<dim3(N_NODES / 8), dim3(256), 0, stream>>>(feat, attn_a, v);
  gat_broadcast<<<dim3(N_NODES / 32, N_NODES), dim3(256), 0, stream>>>(adj, v, out);
}